// Shuffler_19018115187268
// MI455X (gfx1250) — compile-verified
//
#include <hip/hip_runtime.h>
#include <hip/hip_bf16.h>

// Per-row random-prefix shuffle (EGG-style) for B=8192 rows, L=512.
// Memory-bound (~96 MB total traffic -> ~4.5 us at 23.3 TB/s). No matmul ->
// no WMMA; the CDNA5 win here is async global->LDS staging + LDS sort.
//
// One 256-thread (8-wave, wave32) workgroup per row:
//   1) async-copy the int32 message row into LDS (GLOBAL_LOAD_ASYNC_TO_LDS_B64)
//   2) ds_min atomic to find the first EOS -> len
//   3) Threefry-2x32-20 (JAX PRNG, key=(0,42), JAX counter split) per position
//   4) pack (sortable keybits << 32) | pos, 512-wide bitonic sort in LDS
//   5) gather messages through the permutation; B128 pass-through for
//      probs/entropies.

#define L_DIM   512
#define THREADS 256

typedef int v2i32 __attribute__((ext_vector_type(2)));
typedef __attribute__((address_space(1))) v2i32 gv2i32;  // global 2xi32
typedef __attribute__((address_space(3))) v2i32 lv2i32;  // LDS 2xi32

__device__ __forceinline__ unsigned rotl32(unsigned v, unsigned n) {
  return (v << n) | (v >> (32u - n));
}

// Threefry-2x32, 20 rounds; key = jax.random.key(42) -> (k0,k1) = (0,42).
__device__ __forceinline__ void threefry2x32(unsigned c0, unsigned c1,
                                             unsigned& o0, unsigned& o1) {
  const unsigned k0 = 0u, k1 = 42u;
  const unsigned ks2 = k0 ^ k1 ^ 0x1BD11BDAu;
  const unsigned ks[3] = {k0, k1, ks2};
  unsigned x0 = c0 + ks[0];
  unsigned x1 = c1 + ks[1];
  const unsigned R0[4] = {13u, 15u, 26u, 6u};
  const unsigned R1[4] = {17u, 29u, 16u, 24u};
#pragma unroll
  for (int i = 0; i < 5; ++i) {
#pragma unroll
    for (int j = 0; j < 4; ++j) {
      const unsigned r = (i & 1) ? R1[j] : R0[j];
      x0 += x1;
      x1 = rotl32(x1, r);
      x1 ^= x0;
    }
    x0 += ks[(i + 1) % 3];
    x1 += ks[(i + 2) % 3] + (unsigned)(i + 1);
  }
  o0 = x0;
  o1 = x1;
}

__global__ __launch_bounds__(THREADS) void shuffler_rows_kernel(
    const int* __restrict__ messages, const float* __restrict__ probs,
    const float* __restrict__ ents, float* __restrict__ out, int B) {
  __shared__ __align__(16) int s_msg[L_DIM];
  __shared__ __align__(16) unsigned long long s_key[L_DIM];
  __shared__ int s_first_zero;

  const int tid = threadIdx.x;
  const int row = blockIdx.x;
  const long long base = (long long)row * L_DIM;
  const int* msg_row = messages + base;

  if (tid == 0) s_first_zero = L_DIM;

  // ---- Stage message row into LDS (CDNA5 async path; 8 B per lane) ----
#if __has_builtin(__builtin_amdgcn_global_load_async_to_lds_b64)
  __builtin_amdgcn_global_load_async_to_lds_b64(
      (gv2i32*)(msg_row + 2 * tid), (lv2i32*)&s_msg[2 * tid], /*offset=*/0,
      /*cpol=*/0);
#if __has_builtin(__builtin_amdgcn_s_wait_asynccnt)
  __builtin_amdgcn_s_wait_asynccnt(0);
#else
  asm volatile("s_wait_asynccnt 0" ::: "memory");
#endif
#else
  {
    const int2 v = *(const int2*)(msg_row + 2 * tid);
    s_msg[2 * tid + 0] = v.x;
    s_msg[2 * tid + 1] = v.y;
  }
#endif
  __syncthreads();

  // ---- find_lengths: first index of token 0 (ds_min), len = min(fz+1, L) ----
#pragma unroll
  for (int e = 0; e < 2; ++e) {
    const int p = tid + e * THREADS;
    if (s_msg[p] == 0) atomicMin(&s_first_zero, p);
  }
  __syncthreads();
  const int len = min(s_first_zero + 1, L_DIM);

  // ---- per-position sort keys (JAX threefry uniform; suffix = 1+pos) ----
  const unsigned n_half = ((unsigned)B * (unsigned)L_DIM) >> 1;
#pragma unroll
  for (int e = 0; e < 2; ++e) {
    const int p = tid + e * THREADS;
    const unsigned g = (unsigned)row * (unsigned)L_DIM + (unsigned)p;
    unsigned o0, o1, bits;
    if (g < n_half) {
      threefry2x32(g, g + n_half, o0, o1);
      bits = o0;
    } else {
      threefry2x32(g - n_half, g, o0, o1);
      bits = o1;
    }
    const float u = __uint_as_float((bits >> 9) | 0x3f800000u) - 1.0f;
    const float kf = (p < len) ? u : (1.0f + (float)p);
    // all keys are positive floats -> raw bits are order-preserving
    const unsigned kb = __float_as_uint(kf);
    s_key[p] = ((unsigned long long)kb << 32) | (unsigned)p;
  }
  __syncthreads();

  // ---- bitonic sort of 512 packed (key|idx) u64s in LDS ----
  for (unsigned k = 2; k <= (unsigned)L_DIM; k <<= 1) {
    for (unsigned j = k >> 1; j > 0; j >>= 1) {
#pragma unroll
      for (int e = 0; e < 2; ++e) {
        const unsigned i = (unsigned)tid + (unsigned)(e * THREADS);
        const unsigned ixj = i ^ j;
        if (ixj > i) {
          const unsigned long long a = s_key[i];
          const unsigned long long b = s_key[ixj];
          const bool up = ((i & k) == 0u);
          if ((a > b) == up) {
            s_key[i] = b;
            s_key[ixj] = a;
          }
        }
      }
      __syncthreads();
    }
  }

  // ---- gather through perm; shuffled[p] = messages[perm[p]] ----
  float* out_shuf = out;
  float* out_probs = out + (long long)B * L_DIM;
  float* out_ents = out + 2ll * (long long)B * L_DIM;
#pragma unroll
  for (int e = 0; e < 2; ++e) {
    const int p = tid + e * THREADS;
    const unsigned src = (unsigned)(s_key[p] & 0xFFFFFFFFull);
    out_shuf[base + p] = (float)s_msg[src];
  }

  // ---- pass-through copies (16 B/lane; 128 chunks per array per row) ----
  const float4* pr = (const float4*)(probs + base);
  const float4* en = (const float4*)(ents + base);
  float4* opr = (float4*)(out_probs + base);
  float4* oen = (float4*)(out_ents + base);
  if (tid < 128) {
    opr[tid] = pr[tid];
  } else {
    oen[tid - 128] = en[tid - 128];
  }
}

extern "C" void kernel_launch(void* const* d_in, const int* in_sizes, int n_in,
                              void* d_out, int out_size, void* d_ws,
                              size_t ws_size, hipStream_t stream) {
  (void)n_in;
  (void)out_size;
  (void)d_ws;
  (void)ws_size;
  const int* messages = (const int*)d_in[0];
  const float* probs = (const float*)d_in[1];
  const float* ents = (const float*)d_in[2];
  float* out = (float*)d_out;

  const int B = in_sizes[0] / L_DIM;  // 8192
  dim3 grid((unsigned)B), block(THREADS);
  hipLaunchKernelGGL(shuffler_rows_kernel, grid, block, 0, stream, messages,
                     probs, ents, out, B);
}